// LSAPPNPL_mag_6519760355655
// MI455X (gfx1250) — compile-verified
//
#include <hip/hip_runtime.h>
#include <math.h>

// ---------- problem constants ----------
#define NN      500000
#define FDIM    128
#define HDIM    128
#define BB      32768
#define KHOP    4
#define EDG     524288            // 16 * 32768, = 2^19
#define OUTD    8
#define MTOT    (BB + KHOP * EDG) // 2,129,920 rows through the MLP
#define NTILES  (MTOT / 16)       // 133,120 (exact)
#define XS      132               // padded row stride (floats) -> bank-conflict-free col reads
#define PWF     (16 * XS + 64)    // per-wave LDS floats (tile + metadata), 16B-aligned

typedef __attribute__((ext_vector_type(2))) float v2f;
typedef __attribute__((ext_vector_type(8))) float v8f;

__device__ __forceinline__ void wait_ds() {
    asm volatile("s_wait_dscnt 0x0" ::: "memory");
}

// ---------- kernel 1: zero aggx + dsum slot ----------
__global__ void lsappnp_zero(float* __restrict__ ws) {
    size_t i = (size_t)blockIdx.x * blockDim.x + threadIdx.x;
    if (i == 0) ws[0] = 0.0f;
    float* aggx = ws + 64;
    if (i < (size_t)BB * HDIM) aggx[i] = 0.0f;
}

// ---------- kernel 2: dsum = sum(degree[batch]) ----------
__global__ void lsappnp_dsum(const float* __restrict__ degree,
                             const int* __restrict__ batch,
                             float* __restrict__ ws) {
    __shared__ float red[256];
    float s = 0.0f;
    for (int i = blockIdx.x * blockDim.x + threadIdx.x; i < BB;
         i += gridDim.x * blockDim.x)
        s += degree[batch[i]];
    red[threadIdx.x] = s;
    __syncthreads();
    for (int off = 128; off > 0; off >>= 1) {
        if (threadIdx.x < off) red[threadIdx.x] += red[threadIdx.x + off];
        __syncthreads();
    }
    if (threadIdx.x == 0) atomicAdd(ws, red[0]);
}

// ---------- kernel 3: gathered 2-layer MLP via f32 WMMA + segment atomics ----------
__global__ void __launch_bounds__(256, 1)
lsappnp_mlp(const float* __restrict__ x,
            const float* __restrict__ W0, const float* __restrict__ b0,
            const float* __restrict__ W1, const float* __restrict__ b1,
            const float* __restrict__ att,
            const float* __restrict__ degree,
            const int* __restrict__ batch,
            const int* __restrict__ heads,   // [K*E] flat
            const int* __restrict__ ends,    // [K*E] flat
            const float* __restrict__ dsum_p,
            float* __restrict__ aggx) {
    extern __shared__ float smem[];
    // W packed as (K even, K+1) float2 pairs: one ds_load_b64 per B-fragment
    v2f*   W0p   = (v2f*)smem;                 // 64*128 v2f = 64 KB
    v2f*   W1p   = W0p + 64 * 128;             // 64 KB
    float* b0s   = (float*)(W1p + 64 * 128);   // 128
    float* b1s   = b0s + 128;                  // 128
    float* Xbase = b1s + 128;                  // 8 waves * PWF floats

    const int tid  = threadIdx.x;
    const int wave = tid >> 5;
    const int lane = tid & 31;
    const int hi   = lane >> 4;   // lane half (WMMA layout split)
    const int lm   = lane & 15;

    // cooperative weight repack + biases
    for (int i = tid; i < 64 * 128; i += 256) {
        int k2 = i >> 7, n = i & 127;
        v2f w0; w0[0] = W0[(2 * k2) * 128 + n]; w0[1] = W0[(2 * k2 + 1) * 128 + n];
        v2f w1; w1[0] = W1[(2 * k2) * 128 + n]; w1[1] = W1[(2 * k2 + 1) * 128 + n];
        W0p[i] = w0;
        W1p[i] = w1;
    }
    if (tid < 128) { b0s[tid] = b0[tid]; b1s[tid] = b1[tid]; }
    __syncthreads();

    const float dsum   = dsum_p[0];
    const float inv_ne = 1.0f / (float)EDG;

    float* Xs    = Xbase + wave * PWF;          // 16 x 132 staging tile
    int*   metaS = (int*)(Xs + 16 * XS);        // src node ids [16]
    int*   metaT = metaS + 16;                  // target rows  [16]
    float* metaF = (float*)(metaT + 16);        // row scales   [16]

    for (int tile = blockIdx.x * 8 + wave; tile < NTILES; tile += gridDim.x * 8) {
        // ---- per-row metadata (lanes 0..15) ----
        if (lane < 16) {
            long r = (long)tile * 16 + lane;
            int src, tgt; float sc;
            if (r < BB) {
                tgt = (int)r;
                src = batch[r];
                sc  = att[0];
            } else {
                long idx = r - BB;                 // flat [K*E] index
                int  k   = (int)(idx >> 19);       // E = 2^19
                tgt = heads[idx];
                src = ends[idx];
                float db = degree[batch[tgt]];
                float de = degree[src];
                sc = att[k + 1] * rsqrtf(db) * rsqrtf(de) * dsum * inv_ne;
            }
            metaS[lane] = src; metaT[lane] = tgt; metaF[lane] = sc;
        }
        wait_ds();

        // ---- stage 16 x-rows, coalesced float4 per lane ----
        for (int rrow = 0; rrow < 16; ++rrow) {
            int src = metaS[rrow];
            float4 v = ((const float4*)(x + (size_t)src * FDIM))[lane];
            ((float4*)(Xs + rrow * XS))[lane] = v;
        }
        wait_ds();

        // ---- layer 0: h = relu(X @ W0 + b0), 16x128 tile via WMMA f32 16x16x4 ----
        v8f acc[8];
#pragma unroll
        for (int nt = 0; nt < 8; ++nt) {
            float bv = b0s[nt * 16 + lm];
            v8f c;
#pragma unroll
            for (int i = 0; i < 8; ++i) c[i] = bv;
            acc[nt] = c;
        }
        for (int kt = 0; kt < 32; ++kt) {
            int kc = kt * 4 + hi * 2;                       // even
            v2f a  = *(const v2f*)(Xs + lm * XS + kc);      // A: M=lm, K=kc,kc+1
            int k2 = kc >> 1;
#pragma unroll
            for (int nt = 0; nt < 8; ++nt) {
                v2f bfr = W0p[k2 * 128 + nt * 16 + lm];     // B: K rows kc,kc+1, N col
                acc[nt] = __builtin_amdgcn_wmma_f32_16x16x4_f32(
                    false, a, false, bfr, (short)0, acc[nt], false, false);
            }
        }
        // relu -> reuse Xs as H tile (same-wave LDS ops are in order)
#pragma unroll
        for (int nt = 0; nt < 8; ++nt)
#pragma unroll
            for (int v = 0; v < 8; ++v)
                Xs[(v + 8 * hi) * XS + nt * 16 + lm] = fmaxf(acc[nt][v], 0.0f);
        wait_ds();

        // ---- layer 1: y = H @ W1 + b1 ----
        v8f acc2[8];
#pragma unroll
        for (int nt = 0; nt < 8; ++nt) {
            float bv = b1s[nt * 16 + lm];
            v8f c;
#pragma unroll
            for (int i = 0; i < 8; ++i) c[i] = bv;
            acc2[nt] = c;
        }
        for (int kt = 0; kt < 32; ++kt) {
            int kc = kt * 4 + hi * 2;
            v2f a  = *(const v2f*)(Xs + lm * XS + kc);
            int k2 = kc >> 1;
#pragma unroll
            for (int nt = 0; nt < 8; ++nt) {
                v2f bfr = W1p[k2 * 128 + nt * 16 + lm];
                acc2[nt] = __builtin_amdgcn_wmma_f32_16x16x4_f32(
                    false, a, false, bfr, (short)0, acc2[nt], false, false);
            }
        }

        // ---- scale per row, segment-sum via f32 atomics into aggx ----
#pragma unroll
        for (int v = 0; v < 8; ++v) {
            int   tg = metaT[v + 8 * hi];
            float sc = metaF[v + 8 * hi];
            float* base = aggx + (size_t)tg * HDIM + lm;
#pragma unroll
            for (int nt = 0; nt < 8; ++nt)
                atomicAdd(base + nt * 16, acc2[nt][v] * sc);
        }
    }
}

// ---------- kernel 4: out = log_softmax(relu(aggx) @ W2 + b2) ----------
__global__ void lsappnp_head(const float* __restrict__ aggx,
                             const float* __restrict__ W2,
                             const float* __restrict__ b2,
                             float* __restrict__ out) {
    int row  = blockIdx.x * 8 + (threadIdx.x >> 5);
    int lane = threadIdx.x & 31;
    if (row >= BB) return;
    float p[8] = {0, 0, 0, 0, 0, 0, 0, 0};
    for (int c = lane; c < HDIM; c += 32) {
        float a = fmaxf(aggx[(size_t)row * HDIM + c], 0.0f);
        const float4* wp = (const float4*)(W2 + c * OUTD);
        float4 w0 = wp[0], w1 = wp[1];
        p[0] += a * w0.x; p[1] += a * w0.y; p[2] += a * w0.z; p[3] += a * w0.w;
        p[4] += a * w1.x; p[5] += a * w1.y; p[6] += a * w1.z; p[7] += a * w1.w;
    }
#pragma unroll
    for (int off = 16; off > 0; off >>= 1)
#pragma unroll
        for (int j = 0; j < 8; ++j) p[j] += __shfl_down(p[j], off);
    if (lane == 0) {
#pragma unroll
        for (int j = 0; j < 8; ++j) p[j] += b2[j];
        float m = p[0];
#pragma unroll
        for (int j = 1; j < 8; ++j) m = fmaxf(m, p[j]);
        float s = 0.0f;
#pragma unroll
        for (int j = 0; j < 8; ++j) s += __expf(p[j] - m);
        float lse = __logf(s);
#pragma unroll
        for (int j = 0; j < 8; ++j) out[(size_t)row * OUTD + j] = p[j] - m - lse;
    }
}

extern "C" void kernel_launch(void* const* d_in, const int* in_sizes, int n_in,
                              void* d_out, int out_size, void* d_ws, size_t ws_size,
                              hipStream_t stream) {
    const float* x      = (const float*)d_in[0];
    const float* W0     = (const float*)d_in[1];
    const float* b0     = (const float*)d_in[2];
    const float* W1     = (const float*)d_in[3];
    const float* b1     = (const float*)d_in[4];
    const float* W2     = (const float*)d_in[5];
    const float* b2     = (const float*)d_in[6];
    const float* att    = (const float*)d_in[7];
    const float* degree = (const float*)d_in[8];
    const int*   batch  = (const int*)d_in[9];
    const int*   heads  = (const int*)d_in[10];
    const int*   ends   = (const int*)d_in[11];
    float*       out    = (float*)d_out;
    float*       ws     = (float*)d_ws;   // [0]=dsum, +64: aggx [B,H]
    float*       aggx   = ws + 64;

    lsappnp_zero<<<(BB * HDIM + 255) / 256, 256, 0, stream>>>(ws);
    lsappnp_dsum<<<64, 256, 0, stream>>>(degree, batch, ws);

    size_t shmem = (size_t)2 * 64 * 128 * sizeof(v2f)   // W0p + W1p
                 + 256 * sizeof(float)                  // b0s + b1s
                 + (size_t)8 * PWF * sizeof(float);     // per-wave tiles
    lsappnp_mlp<<<1280, 256, shmem, stream>>>(x, W0, b0, W1, b1, att, degree,
                                              batch, heads, ends, ws, aggx);

    lsappnp_head<<<BB / 8, 256, 0, stream>>>(aggx, W2, b2, out);
}